// Module_58901181497370
// MI455X (gfx1250) — compile-verified
//
#include <hip/hip_runtime.h>
#include <hip/hip_bf16.h>
#include <math.h>

#define B_DIM 512
#define L_DIM 1024
#define D_DIM 256
#define SCALE 0.0625f   // 1/sqrt(256)/TEMP

typedef float v2f __attribute__((ext_vector_type(2)));
typedef float v8f __attribute__((ext_vector_type(8)));
typedef int   v4i __attribute__((__vector_size__(16)));

// ---- CDNA5 async-to-LDS helpers (builtin if present, else inline asm) ------
#if defined(__has_builtin) && __has_builtin(__builtin_amdgcn_s_wait_asynccnt)
#define WAIT_ASYNC(n) __builtin_amdgcn_s_wait_asynccnt(n)
#else
#define WAIT_ASYNC(n) asm volatile("s_wait_asynccnt " #n ::: "memory")
#endif

__device__ __forceinline__ void async_ld16(const float* g, float* l)
{
#if defined(__has_builtin) && __has_builtin(__builtin_amdgcn_global_load_async_to_lds_b128)
  __builtin_amdgcn_global_load_async_to_lds_b128(
      (__attribute__((address_space(1))) v4i*)g,
      (__attribute__((address_space(3))) v4i*)l, 0, 0);
#else
  unsigned loff = (unsigned)(unsigned long long)l;   // LDS byte offset (low 32 bits)
  asm volatile("global_load_async_to_lds_b128 %0, %1, off"
               :: "v"(loff), "v"(g) : "memory");
#endif
}

// ---------------------------------------------------------------------------
// C[M,N] = A[M,K] * op(B) (+ bias), op(B)=B^T when TRANSB (B stored [N,K]),
// else B stored [K,N].  One wave per 16x16 output tile, f32 WMMA 16x16x4.
// ---------------------------------------------------------------------------
template <bool TRANSB, bool HASBIAS>
__global__ __launch_bounds__(128) void gemm_f32_wmma(
    const float* __restrict__ A, const float* __restrict__ Bm,
    const float* __restrict__ bias, float* __restrict__ C,
    int M, int N, int Kd)
{
  const int lane  = threadIdx.x & 31;
  const int wave  = threadIdx.y;
  const int tileN = blockIdx.x;
  const int tileM = blockIdx.y * 4 + wave;
  const int half  = lane >> 4;
  const int mr = tileM * 16 + (lane & 15);
  const int nc = tileN * 16 + (lane & 15);
  const int kh = half * 2;

  v8f acc = {0.f, 0.f, 0.f, 0.f, 0.f, 0.f, 0.f, 0.f};

  for (int kk = 0; kk < Kd; kk += 4) {
    const int k0 = kk + kh;
    v2f a, b;
    a.x = A[(size_t)mr * Kd + k0];
    a.y = A[(size_t)mr * Kd + k0 + 1];
    if (TRANSB) {
      b.x = Bm[(size_t)nc * Kd + k0];
      b.y = Bm[(size_t)nc * Kd + k0 + 1];
    } else {
      b.x = Bm[(size_t)k0 * N + nc];
      b.y = Bm[(size_t)(k0 + 1) * N + nc];
    }
    acc = __builtin_amdgcn_wmma_f32_16x16x4_f32(false, a, false, b,
                                                (short)0, acc, false, false);
  }

  const float bb = HASBIAS ? bias[nc] : 0.f;
#pragma unroll
  for (int v = 0; v < 8; ++v) {
    const int row = tileM * 16 + v + half * 8;
    C[(size_t)row * N + nc] = acc[v] + bb;
  }
}

// ---------------------------------------------------------------------------
// c[b] = q[b,:] . bk   (one 256-thread block per b)
// ---------------------------------------------------------------------------
__global__ __launch_bounds__(256) void dot_bk_kernel(
    const float* __restrict__ q, const float* __restrict__ bk,
    float* __restrict__ c)
{
  __shared__ float red[8];
  const int b = blockIdx.x, tid = threadIdx.x;
  float p = q[(size_t)b * D_DIM + tid] * bk[tid];
#pragma unroll
  for (int o = 16; o >= 1; o >>= 1) p += __shfl_xor(p, o, 32);
  if ((tid & 31) == 0) red[tid >> 5] = p;
  __syncthreads();
  if (tid == 0) {
    float s = 0.f;
#pragma unroll
    for (int i = 0; i < 8; ++i) s += red[i];
    c[b] = s;
  }
}

// ---------------------------------------------------------------------------
// Streaming attention with async double-buffered K tiles:
//   scores = (qk_b . K[b,l] + c_b)*SCALE, masked softmax, weights -> d_out,
//   u_b = sum_l w_l * V[b,l] -> workspace.
// One 256-thread block per batch row.
// ---------------------------------------------------------------------------
__global__ __launch_bounds__(256) void attn_stream_kernel(
    const float* __restrict__ K, const float* __restrict__ V,
    const unsigned char* __restrict__ padding,
    const float* __restrict__ qk, const float* __restrict__ cofs,
    float* __restrict__ weights_out, float* __restrict__ u_out)
{
  constexpr int TILE  = 16;                 // K rows per buffer
  constexpr int NTILE = L_DIM / TILE;       // 64
  __shared__ __align__(16) float s_tile[2][TILE * D_DIM];  // 2 x 16 KB
  __shared__ float s_qk[D_DIM];
  __shared__ float s_scores[L_DIM];
  __shared__ float s_red[8];

  const int b = blockIdx.x;
  const int tid = threadIdx.x;

  s_qk[tid] = qk[(size_t)b * D_DIM + tid];

  const float cb = cofs[b];
  const float* Kb = K + (size_t)b * L_DIM * D_DIM;

  // per-thread async fill of one 16x256 tile: 4 x b128 each (1024 float4)
  auto issue = [&](int t, int buf) {
    const float* src = Kb + (size_t)t * TILE * D_DIM;
#pragma unroll
    for (int i = 0; i < 4; ++i) {
      const int idx = tid + 256 * i;      // float4 index
      async_ld16(src + idx * 4, &s_tile[buf][idx * 4]);
    }
  };

  // ---- phase 1: scores over L, async double-buffered through LDS ----
  issue(0, 0);
  __syncthreads();                          // s_qk visible too

  for (int t = 0; t < NTILE; ++t) {
    const int cur = t & 1;
    if (t + 1 < NTILE) {
      issue(t + 1, cur ^ 1);
      WAIT_ASYNC(4);                        // tile t's 4 requests complete
    } else {
      WAIT_ASYNC(0);
    }
    __syncthreads();                        // all waves' tile-t data in LDS

    const int g = tid >> 4;                 // row in tile (0..15)
    const int j = tid & 15;                 // 16 threads per row
    const float* row = &s_tile[cur][g * D_DIM];
    float p = 0.f;
#pragma unroll
    for (int k = 0; k < 16; ++k) {
      const int dd = j * 16 + ((k + j) & 15);   // bank-conflict-free swizzle
      p += row[dd] * s_qk[dd];
    }
    p += __shfl_xor(p, 8, 32);
    p += __shfl_xor(p, 4, 32);
    p += __shfl_xor(p, 2, 32);
    p += __shfl_xor(p, 1, 32);
    if (j == 0) s_scores[t * TILE + g] = p;
    __syncthreads();                        // compute done before buffer reuse
  }

  // ---- phase 2: masked softmax over L (each thread owns 4 l's) ----
  const unsigned char* pad = padding + (size_t)b * L_DIM;
  float sc[4];
  float lmax = -INFINITY;
#pragma unroll
  for (int i = 0; i < 4; ++i) {
    const int l = tid + 256 * i;
    float s = (s_scores[l] + cb) * SCALE;
    if (pad[l]) s = -INFINITY;
    sc[i] = s;
    lmax = fmaxf(lmax, s);
  }
#pragma unroll
  for (int o = 16; o >= 1; o >>= 1) lmax = fmaxf(lmax, __shfl_xor(lmax, o, 32));
  if ((tid & 31) == 0) s_red[tid >> 5] = lmax;
  __syncthreads();
  float bmax = s_red[0];
#pragma unroll
  for (int i = 1; i < 8; ++i) bmax = fmaxf(bmax, s_red[i]);
  __syncthreads();

  float lsum = 0.f;
#pragma unroll
  for (int i = 0; i < 4; ++i) {
    sc[i] = __expf(sc[i] - bmax);           // exp(-inf)=0 handles mask
    lsum += sc[i];
  }
#pragma unroll
  for (int o = 16; o >= 1; o >>= 1) lsum += __shfl_xor(lsum, o, 32);
  if ((tid & 31) == 0) s_red[tid >> 5] = lsum;
  __syncthreads();
  float bsum = 0.f;
#pragma unroll
  for (int i = 0; i < 8; ++i) bsum += s_red[i];
  const float inv = 1.0f / bsum;

#pragma unroll
  for (int i = 0; i < 4; ++i) {
    const int l = tid + 256 * i;
    const float w = sc[i] * inv;
    s_scores[l] = w;
    weights_out[(size_t)b * L_DIM + l] = w;
  }
  __syncthreads();                          // all waves past phases 1-2

  // ---- phase 3: u[d] = sum_l w[l] * V[b,l,d], float4 per thread ----
  const int q4 = (tid & 63) * 4;            // d quad base
  const int gl = tid >> 6;                  // l-phase 0..3
  const float* Vb = V + (size_t)b * L_DIM * D_DIM;
  float4 acc4 = {0.f, 0.f, 0.f, 0.f};
  for (int l = gl; l < L_DIM; l += 4) {
    const int lp = (l + 16 < L_DIM) ? (l + 16) : (L_DIM - 1);
    __builtin_prefetch(Vb + (size_t)lp * D_DIM + q4, 0, 0);
    const float w = s_scores[l];
    const float4 vv = *(const float4*)(Vb + (size_t)l * D_DIM + q4);
    acc4.x = fmaf(w, vv.x, acc4.x);
    acc4.y = fmaf(w, vv.y, acc4.y);
    acc4.z = fmaf(w, vv.z, acc4.z);
    acc4.w = fmaf(w, vv.w, acc4.w);
  }
  float* s_part = &s_tile[0][0];            // reuse tile buffer (4 KB needed)
  *(float4*)(s_part + gl * D_DIM + q4) = acc4;
  __syncthreads();
  const float u = s_part[tid] + s_part[D_DIM + tid] +
                  s_part[2 * D_DIM + tid] + s_part[3 * D_DIM + tid];
  u_out[(size_t)b * D_DIM + tid] = u;
}

// ---------------------------------------------------------------------------
extern "C" void kernel_launch(void* const* d_in, const int* in_sizes, int n_in,
                              void* d_out, int out_size, void* d_ws, size_t ws_size,
                              hipStream_t stream)
{
  const float* Q  = (const float*)d_in[0];
  const float* K  = (const float*)d_in[1];
  const float* V  = (const float*)d_in[2];
  const unsigned char* pad = (const unsigned char*)d_in[3];
  const float* Wq = (const float*)d_in[4];
  const float* bq = (const float*)d_in[5];
  const float* Wk = (const float*)d_in[6];
  const float* bk = (const float*)d_in[7];
  const float* Wv = (const float*)d_in[8];
  const float* bv = (const float*)d_in[9];

  float* ctx_out = (float*)d_out;                        // B*D context
  float* w_out   = ctx_out + (size_t)B_DIM * D_DIM;      // B*L weights

  float* ws    = (float*)d_ws;
  float* q_ws  = ws;                                     // B*D
  float* qk_ws = q_ws + (size_t)B_DIM * D_DIM;           // B*D
  float* c_ws  = qk_ws + (size_t)B_DIM * D_DIM;          // B
  float* u_ws  = c_ws + B_DIM;                           // B*D

  dim3 gblock(32, 4);
  dim3 ggrid(D_DIM / 16, B_DIM / 64);

  // q = Q @ Wq^T + bq
  gemm_f32_wmma<true, true><<<ggrid, gblock, 0, stream>>>(
      Q, Wq, bq, q_ws, B_DIM, D_DIM, D_DIM);
  // c_b = q_b . bk
  dot_bk_kernel<<<B_DIM, 256, 0, stream>>>(q_ws, bk, c_ws);
  // qk = q @ Wk
  gemm_f32_wmma<false, false><<<ggrid, gblock, 0, stream>>>(
      q_ws, Wk, nullptr, qk_ws, B_DIM, D_DIM, D_DIM);
  // streaming scores + softmax + weighted V accumulation
  attn_stream_kernel<<<B_DIM, 256, 0, stream>>>(
      K, V, pad, qk_ws, c_ws, w_out, u_ws);
  // context = u @ Wv^T + bv
  gemm_f32_wmma<true, true><<<ggrid, gblock, 0, stream>>>(
      u_ws, Wv, bv, ctx_out, B_DIM, D_DIM, D_DIM);
}